// MultiTimeAttention_23132693856878
// MI455X (gfx1250) — compile-verified
//
#include <hip/hip_runtime.h>
#include <hip/hip_bf16.h>
#include <math.h>

// ---------------------------------------------------------------------------
// MultiTimeAttention for MI455X (gfx1250, wave32, WMMA).
// Sizes: B=4, LQ=128, LK=512, E=128, H=4, DK=32, D=64, NH=128.
// GEMMs on v_wmma_f32_16x16x4_f32 (fp32 matrix pipe, full reference precision).
// Softmax stage uses global_load_async_to_lds_b128 (ASYNCcnt path).
// ---------------------------------------------------------------------------

typedef __attribute__((ext_vector_type(2))) float v2f;
typedef __attribute__((ext_vector_type(8))) float v8f;

#define Bn   4
#define LQ   128
#define LK   512
#define E    128
#define Hh   4
#define DK   32   // E / H
#define Dd   64
#define NH   128
#define HD   (Hh * Dd)   // 256

// f32 WMMA 16x16x4 lane mapping (wave32):
//   A (16x4): lane L holds row (L&15), K = 2*(L>>4) + {0,1} in 2 VGPRs
//   B (4x16): lane L holds col (L&15), K = 2*(L>>4) + {0,1} in 2 VGPRs
//   C/D (16x16, 8 VGPRs): vgpr i -> M = i + 8*(L>>4), N = (L&15)

// ---------------------------------------------------------------------------
// Kernel 1/2: projection  out_headsplit[b][h][row][dk] = in[row][:]@W + bias
//   in: (B*ROWS) x E row-major, W: E x E row-major, bias: E
//   ROWS is compile-time (power of two) so the batch split is a shift.
//   grid.x = totalRows/16, grid.y = E/16, block = 32 (one wave per tile)
// ---------------------------------------------------------------------------
template <int ROWS>
__global__ void proj_headsplit_kernel(const float* __restrict__ in,
                                      const float* __restrict__ W,
                                      const float* __restrict__ bias,
                                      float* __restrict__ out) {
  const int m0   = blockIdx.x << 4;
  const int n0   = blockIdx.y << 4;
  const int lane = threadIdx.x;
  const int r    = lane & 15;        // A row / B col within tile
  const int kh2  = (lane >> 4) << 1; // K sub-offset: 0 or 2

  v8f c = {};
  for (int kk = 0; kk < E; kk += 4) {
    const int k0 = kk + kh2;
    v2f a, b;
    a.x = in[(m0 + r) * E + k0];
    a.y = in[(m0 + r) * E + k0 + 1];
    b.x = W[(k0    ) * E + (n0 + r)];
    b.y = W[(k0 + 1) * E + (n0 + r)];
    c = __builtin_amdgcn_wmma_f32_16x16x4_f32(false, a, false, b,
                                              (short)0, c, false, false);
  }

  const int mofs = (lane >> 4) << 3;          // 0 or 8
  const int j    = n0 + r;                    // full E column
  const float bj = bias[j];
  const int h    = j >> 5;                    // DK = 32
  const int dk   = j & 31;
  for (int i = 0; i < 8; ++i) {
    const int grow = m0 + mofs + i;           // row in concatenated B*ROWS
    const int bb   = grow / ROWS;             // compile-time -> shift
    const int rr   = grow & (ROWS - 1);
    out[((bb * Hh + h) * ROWS + rr) * DK + dk] = c[i] + bj;
  }
}

// ---------------------------------------------------------------------------
// Kernel 3: scores S[bh][q][k] = (Qh[q][:] . Kh[k][:]) / sqrt(DK)
//   Qh: [bh][LQ][DK], Kh: [bh][LK][DK]   (B acts as Kh^T: B[k'][n] = Kh[n][k'])
//   grid = (LQ/16, LK/16, B*H), block = 32
// ---------------------------------------------------------------------------
__global__ void scores_kernel(const float* __restrict__ Qh,
                              const float* __restrict__ Kh,
                              float* __restrict__ S) {
  const int bh   = blockIdx.z;
  const int m0   = blockIdx.x << 4;
  const int n0   = blockIdx.y << 4;
  const int lane = threadIdx.x;
  const int r    = lane & 15;
  const int kh2  = (lane >> 4) << 1;

  const float* q  = Qh + bh * LQ * DK;
  const float* kp = Kh + bh * LK * DK;

  v8f c = {};
  for (int kk = 0; kk < DK; kk += 4) {
    const int k0 = kk + kh2;
    v2f a, b;
    a.x = q [(m0 + r) * DK + k0];
    a.y = q [(m0 + r) * DK + k0 + 1];
    b.x = kp[(n0 + r) * DK + k0];      // B[k'][n] = Kh[n][k']
    b.y = kp[(n0 + r) * DK + k0 + 1];
    c = __builtin_amdgcn_wmma_f32_16x16x4_f32(false, a, false, b,
                                              (short)0, c, false, false);
  }

  const float scale = 0.17677669529663687f;   // 1/sqrt(32)
  const int mofs = (lane >> 4) << 3;
  for (int i = 0; i < 8; ++i) {
    const int qrow = m0 + mofs + i;
    S[(bh * LQ + qrow) * LK + (n0 + r)] = c[i] * scale;
  }
}

// ---------------------------------------------------------------------------
// Kernel 4: per-channel masked softmax + weighted value sum.
//   For each (b,h,q,d):  s_k = mask[b,k,d] ? S[b,h,q,k] : -1e9
//                        X[b,q,h*64+d] = sum_k softmax_k(s)_k * value[b,k,d]
//   grid = B*LQ blocks (one per (b,q)); block = 256 threads = (h = t>>6, d = t&63).
//   Score rows for all 4 heads staged in 8KB LDS via ASYNC B128 copies
//   (ASYNCcnt, no VGPR bounce); mask/value reads are contiguous across d
//   (coalesced) with 4x h-reuse out of WGP$.
//   Using exactly -1e9 reproduces the reference's all-masked edge case.
// ---------------------------------------------------------------------------
__global__ void attn_softmax_kernel(const float* __restrict__ S,
                                    const int*   __restrict__ mask,
                                    const float* __restrict__ value,
                                    float* __restrict__ X) {
  const int bq = blockIdx.x;
  const int b  = bq >> 7;            // LQ = 128
  const int qi = bq & 127;
  const int t  = threadIdx.x;
  const int h  = t >> 6;
  const int d  = t & 63;

  __shared__ __align__(16) float srow[Hh * LK];    // 8 KB

#if defined(__HIP_DEVICE_COMPILE__)
  // Async global->LDS copy, 16B per lane per issue, tracked on ASYNCcnt.
  // Low 32 bits of a flat pointer to __shared__ carry the wave-relative LDS
  // byte offset (ISA 10.2 aperture rules) -> valid async-LDS dest address.
  for (int idx4 = t * 4; idx4 < Hh * LK; idx4 += 256 * 4) {
    const int hh = idx4 >> 9;        // LK = 512
    const int k  = idx4 & 511;
    const float* gp = S + (((size_t)(b * Hh + hh) * LQ + qi) * LK + k);
    unsigned int       laddr = (unsigned int)(unsigned long long)(&srow[idx4]);
    unsigned long long gaddr = (unsigned long long)gp;
    asm volatile("global_load_async_to_lds_b128 %0, %1, off"
                 :: "v"(laddr), "v"(gaddr) : "memory");
  }
  asm volatile("s_wait_asynccnt 0" ::: "memory");
#else
  for (int idx = t; idx < Hh * LK; idx += 256) {
    const int hh = idx >> 9;
    const int k  = idx & 511;
    srow[idx] = S[((b * Hh + hh) * LQ + qi) * LK + k];
  }
#endif
  __syncthreads();

  const float* sh = srow + (h << 9);
  const int*   mrow = mask  + (size_t)b * LK * Dd + d;   // stride Dd
  const float* vrow = value + (size_t)b * LK * Dd + d;

  // pass 1: masked max (4 independent chains for VALU ILP)
  float mx0 = -3.0e38f, mx1 = -3.0e38f, mx2 = -3.0e38f, mx3 = -3.0e38f;
  for (int k = 0; k < LK; k += 4) {
    const float s0 = mrow[(k + 0) * Dd] ? sh[k + 0] : -1.0e9f;
    const float s1 = mrow[(k + 1) * Dd] ? sh[k + 1] : -1.0e9f;
    const float s2 = mrow[(k + 2) * Dd] ? sh[k + 2] : -1.0e9f;
    const float s3 = mrow[(k + 3) * Dd] ? sh[k + 3] : -1.0e9f;
    mx0 = fmaxf(mx0, s0); mx1 = fmaxf(mx1, s1);
    mx2 = fmaxf(mx2, s2); mx3 = fmaxf(mx3, s3);
  }
  const float mx = fmaxf(fmaxf(mx0, mx1), fmaxf(mx2, mx3));

  // pass 2: exp-sum and weighted accumulation (2 chains each)
  float num0 = 0.0f, num1 = 0.0f, den0 = 0.0f, den1 = 0.0f;
  for (int k = 0; k < LK; k += 2) {
    const float s0 = mrow[(k + 0) * Dd] ? sh[k + 0] : -1.0e9f;
    const float s1 = mrow[(k + 1) * Dd] ? sh[k + 1] : -1.0e9f;
    const float e0 = __expf(s0 - mx);
    const float e1 = __expf(s1 - mx);
    den0 += e0;                      den1 += e1;
    num0 += e0 * vrow[(k + 0) * Dd]; num1 += e1 * vrow[(k + 1) * Dd];
  }
  X[((size_t)b * LQ + qi) * HD + (h << 6) + d] =
      (num0 + num1) / (den0 + den1);
}

// ---------------------------------------------------------------------------
// Kernel 5: out = X @ Wo + bo.   X: (B*LQ) x 256, Wo: 256 x 128.
//   grid = (B*LQ/16, NH/16), block = 32
// ---------------------------------------------------------------------------
__global__ void out_proj_kernel(const float* __restrict__ X,
                                const float* __restrict__ Wo,
                                const float* __restrict__ bo,
                                float* __restrict__ out) {
  const int m0   = blockIdx.x << 4;
  const int n0   = blockIdx.y << 4;
  const int lane = threadIdx.x;
  const int r    = lane & 15;
  const int kh2  = (lane >> 4) << 1;

  v8f c = {};
  for (int kk = 0; kk < HD; kk += 4) {
    const int k0 = kk + kh2;
    v2f a, b;
    a.x = X [(m0 + r) * HD + k0];
    a.y = X [(m0 + r) * HD + k0 + 1];
    b.x = Wo[(k0    ) * NH + (n0 + r)];
    b.y = Wo[(k0 + 1) * NH + (n0 + r)];
    c = __builtin_amdgcn_wmma_f32_16x16x4_f32(false, a, false, b,
                                              (short)0, c, false, false);
  }

  const int mofs = (lane >> 4) << 3;
  const float bj = bo[n0 + r];
  for (int i = 0; i < 8; ++i)
    out[(m0 + mofs + i) * NH + (n0 + r)] = c[i] + bj;
}

// ---------------------------------------------------------------------------
// Launch
// ---------------------------------------------------------------------------
extern "C" void kernel_launch(void* const* d_in, const int* in_sizes, int n_in,
                              void* d_out, int out_size, void* d_ws, size_t ws_size,
                              hipStream_t stream) {
  const float* query = (const float*)d_in[0];   // (4,128,128)
  const float* key   = (const float*)d_in[1];   // (4,512,128)
  const float* value = (const float*)d_in[2];   // (4,512,64)
  const int*   mask  = (const int*)  d_in[3];   // (4,512,64)
  const float* Wq    = (const float*)d_in[4];   // (128,128)
  const float* bq    = (const float*)d_in[5];   // (128)
  const float* Wk    = (const float*)d_in[6];   // (128,128)
  const float* bk    = (const float*)d_in[7];   // (128)
  const float* Wo    = (const float*)d_in[8];   // (256,128)
  const float* bo    = (const float*)d_in[9];   // (128)
  float* out = (float*)d_out;                   // (4,128,128)

  // Workspace carve-up (floats): Qh 64K | Kh 256K | S 1M | X 128K  (~5.75 MB)
  float* Qh = (float*)d_ws;                 // [B][H][LQ][DK]
  float* Kh = Qh + Bn * Hh * LQ * DK;       // [B][H][LK][DK]
  float* S  = Kh + Bn * Hh * LK * DK;       // [B*H][LQ][LK]
  float* X  = S  + Bn * Hh * LQ * LK;       // [B][LQ][H*D]

  // 1) Q projection: 512 rows x 128, head-split output
  proj_headsplit_kernel<LQ><<<dim3((Bn * LQ) / 16, E / 16), 32, 0, stream>>>(
      query, Wq, bq, Qh);
  // 2) K projection: 2048 rows x 128
  proj_headsplit_kernel<LK><<<dim3((Bn * LK) / 16, E / 16), 32, 0, stream>>>(
      key, Wk, bk, Kh);
  // 3) scores per (b,h): 128x512, K=32
  scores_kernel<<<dim3(LQ / 16, LK / 16, Bn * Hh), 32, 0, stream>>>(Qh, Kh, S);
  // 4) per-channel masked softmax + weighted sum (async LDS staging)
  attn_softmax_kernel<<<dim3(Bn * LQ), 256, 0, stream>>>(S, mask, value, X);
  // 5) output projection: 512x128, K=256
  out_proj_kernel<<<dim3((Bn * LQ) / 16, NH / 16), 32, 0, stream>>>(X, Wo, bo, out);
}